// GATModel_78176994722447
// MI455X (gfx1250) — compile-verified
//
#include <hip/hip_runtime.h>

typedef float v2f __attribute__((ext_vector_type(2)));
typedef float v8f __attribute__((ext_vector_type(8)));

// ---------------------------------------------------------------- utilities
__global__ void zero_ws_kernel(unsigned int* p, long nwords) {
  long i = (long)blockIdx.x * blockDim.x + threadIdx.x;
  long stride = (long)gridDim.x * blockDim.x;
  for (; i < nwords; i += stride) p[i] = 0u;
}

// count in-degree (real edges) and accumulate loop_attr sums
__global__ void count_kernel(const int* __restrict__ dst, const float* __restrict__ ea,
                             int* cnt, float* loop_sum, int E) {
  int e = blockIdx.x * blockDim.x + threadIdx.x;
  if (e >= E) return;
  int d = dst[e];
  atomicAdd(&cnt[d], 1);
#pragma unroll
  for (int k = 0; k < 6; k++) atomicAdd(&loop_sum[d * 6 + k], ea[e * 6 + k]);
}

// loop_attr = sum / max(cnt,1)
__global__ void finalize_kernel(const int* cnt, float* loop_attr, int n) {
  int i = blockIdx.x * blockDim.x + threadIdx.x;
  if (i >= n) return;
  int c = cnt[i];
  float inv = 1.0f / (float)(c > 1 ? c : 1);
#pragma unroll
  for (int k = 0; k < 6; k++) loop_attr[i * 6 + k] *= inv;
}

// exclusive scan of (cnt[i]+1) -> rowptr (single block, sequential chunks)
__global__ void scan_kernel(const int* cnt, int* rowptr, int n) {
  __shared__ int buf[256];
  __shared__ int carry;
  int tid = threadIdx.x;
  if (tid == 0) carry = 0;
  __syncthreads();
  for (int base = 0; base < n; base += 256) {
    int i = base + tid;
    int v = (i < n) ? (cnt[i] + 1) : 0;  // +1 self loop
    buf[tid] = v;
    __syncthreads();
    for (int off = 1; off < 256; off <<= 1) {
      int t = (tid >= off) ? buf[tid - off] : 0;
      __syncthreads();
      buf[tid] += t;
      __syncthreads();
    }
    if (i < n) rowptr[i] = carry + buf[tid] - v;
    __syncthreads();
    if (tid == 0) carry += buf[255];
    __syncthreads();
  }
  if (tid == 0) rowptr[n] = carry;
}

__global__ void fill_edges_kernel(const int* __restrict__ src, const int* __restrict__ dst,
                                  const float* __restrict__ ea, const int* __restrict__ rowptr,
                                  int* fill, int* csr_src, float* ea_csr, int E) {
  int e = blockIdx.x * blockDim.x + threadIdx.x;
  if (e >= E) return;
  int d = dst[e];
  int pos = rowptr[d] + atomicAdd(&fill[d], 1);
  csr_src[pos] = src[e];
#pragma unroll
  for (int k = 0; k < 6; k++) ea_csr[pos * 6 + k] = ea[e * 6 + k];
}

__global__ void fill_loops_kernel(const float* __restrict__ loop_attr, const int* __restrict__ rowptr,
                                  int* fill, int* csr_src, float* ea_csr, int n) {
  int i = blockIdx.x * blockDim.x + threadIdx.x;
  if (i >= n) return;
  int pos = rowptr[i] + atomicAdd(&fill[i], 1);
  csr_src[pos] = i;
#pragma unroll
  for (int k = 0; k < 6; k++) ea_csr[pos * 6 + k] = loop_attr[i * 6 + k];
}

// we_ae[l][h][k] = sum_c We_l[k, h*64+c] * a_e_l[h, c]   (layer1: [0..23], layer2: [24..47], layer3: [48..53])
__global__ void weae_kernel(const float* We1, const float* ae1, const float* We2, const float* ae2,
                            const float* We3, const float* ae3, float* we_ae) {
  int t = threadIdx.x;
  if (t < 24) {
    int h = t / 6, k = t % 6;
    float s = 0.f;
    for (int c = 0; c < 64; c++) s += We1[k * 256 + h * 64 + c] * ae1[h * 64 + c];
    we_ae[t] = s;
  } else if (t < 48) {
    int u = t - 24, h = u / 6, k = u % 6;
    float s = 0.f;
    for (int c = 0; c < 64; c++) s += We2[k * 256 + h * 64 + c] * ae2[h * 64 + c];
    we_ae[t] = s;
  } else if (t < 54) {
    int k = t - 48;
    float s = 0.f;
    for (int c = 0; c < 64; c++) s += We3[k * 64 + c] * ae3[c];
    we_ae[t] = s;
  }
}

// ---------------------------------------------------------------- GEMMs
// layer 1: K=7, plain FMA (tiny, memory bound on x)
__global__ void gemm7_kernel(const float* __restrict__ x, const float* __restrict__ W,
                             float* __restrict__ out, int n) {
  int idx = blockIdx.x * blockDim.x + threadIdx.x;
  if (idx >= n * 256) return;
  int row = idx >> 8, col = idx & 255;
  float s = 0.f;
#pragma unroll
  for (int k = 0; k < 7; k++) s += x[row * 7 + k] * W[k * 256 + col];
  out[idx] = s;
}

// layers 2/3: fp32 WMMA 16x16x4. Block = 256 threads = 8 waves; each wave owns a 16x64 tile.
template <int CO>
__global__ void gemm_wmma_kernel(const float* __restrict__ A, const float* __restrict__ W,
                                 float* __restrict__ Out, int nrows) {
  const int K = 256;
  const int WAVES_N = CO / 64;        // 4 (CO=256) or 1 (CO=64)
  const int WAVES_M = 8 / WAVES_N;    // 2 or 8
  int tid = threadIdx.x;
  int lane = tid & 31, wid = tid >> 5;
  int waveM = wid / WAVES_N, waveN = wid % WAVES_N;
  int m0 = blockIdx.x * (WAVES_M * 16) + waveM * 16;
  int n0 = waveN * 64;
  int col16 = lane & 15;
  int mrow = m0 + col16;
  if (mrow >= nrows) mrow = nrows - 1;  // clamp (stores are guarded)
  int kb = (lane >> 4) * 2;             // K-pair selector per half-wave
  int rowbase = m0 + (lane >> 4) * 8;   // D layout: VGPR j -> M = j (lanes 0-15) / j+8 (lanes 16-31)

#if __has_builtin(__builtin_amdgcn_wmma_f32_16x16x4_f32)
  v8f acc[4] = {};
  for (int k = 0; k < K; k += 4) {
    v2f a;
    a.x = A[mrow * K + k + kb];
    a.y = A[mrow * K + k + kb + 1];
#pragma unroll
    for (int t = 0; t < 4; t++) {
      int col = n0 + t * 16 + col16;
      v2f b;
      b.x = W[(k + kb) * CO + col];
      b.y = W[(k + kb + 1) * CO + col];
      acc[t] = __builtin_amdgcn_wmma_f32_16x16x4_f32(false, a, false, b, (short)0, acc[t],
                                                     false, false);
    }
  }
#pragma unroll
  for (int t = 0; t < 4; t++) {
#pragma unroll
    for (int j = 0; j < 8; j++) {
      int row = rowbase + j;
      if (row < nrows) Out[row * CO + n0 + t * 16 + col16] = acc[t][j];
    }
  }
#else
  // fallback (also taken by the host compile pass): same tile, scalar FMA
  for (int t = 0; t < 4; t++) {
    for (int j = 0; j < 8; j++) {
      int row = rowbase + j;
      if (row < nrows) {
        int col = n0 + t * 16 + col16;
        float s = 0.f;
        for (int k = 0; k < K; k++) s += A[row * K + k] * W[k * CO + col];
        Out[row * CO + col] = s;
      }
    }
  }
#endif
}

// ---------------------------------------------------------------- attention
// per-node attention scores: s_score[n,h] = h[n,h,:]·a_s[h], d_score likewise
template <int H>
__global__ void score_kernel(const float* __restrict__ hlin, const float* __restrict__ a_s,
                             const float* __restrict__ a_d, float* s_out, float* d_out, int n) {
  int idx = blockIdx.x * blockDim.x + threadIdx.x;
  if (idx >= n * H) return;
  int node = idx / H, h = idx % H;
  const float* hp = hlin + (long)node * (H * 64) + h * 64;
  const float* ap = a_s + h * 64;
  const float* dp = a_d + h * 64;
  float ss = 0.f, dd = 0.f;
  for (int c = 0; c < 64; c++) {
    float v = hp[c];
    ss += v * ap[c];
    dd += v * dp[c];
  }
  s_out[idx] = ss;
  d_out[idx] = dd;
}

// per-dst-node softmax over CSR segment; wbuf[pos,h] ends as normalized weight
template <int H>
__global__ void attn_kernel(const int* __restrict__ rowptr, const int* __restrict__ csr_src,
                            const float* __restrict__ ea_csr, const float* __restrict__ s_score,
                            const float* __restrict__ d_score, const float* __restrict__ we_ae,
                            float* __restrict__ wbuf) {
  const int T = 64;
  int n = blockIdx.x;
  int tid = threadIdx.x;
  __shared__ float we[H * 6];
  __shared__ float ds[H];
  __shared__ float red[T];
  __shared__ float mh[H], sh[H];
  if (tid < H * 6) we[tid] = we_ae[tid];
  if (tid < H) ds[tid] = d_score[n * H + tid];
  __syncthreads();
  int start = rowptr[n], end = rowptr[n + 1];

  float lmax[H];
#pragma unroll
  for (int h = 0; h < H; h++) lmax[h] = -3.0e38f;
  for (int pos = start + tid; pos < end; pos += T) {
    int s = csr_src[pos];
    float attr[6];
#pragma unroll
    for (int k = 0; k < 6; k++) attr[k] = ea_csr[pos * 6 + k];
#pragma unroll
    for (int h = 0; h < H; h++) {
      float l = s_score[s * H + h] + ds[h];
#pragma unroll
      for (int k = 0; k < 6; k++) l += attr[k] * we[h * 6 + k];
      l = (l > 0.f) ? l : 0.2f * l;  // leaky_relu(0.2)
      wbuf[pos * H + h] = l;
      lmax[h] = fmaxf(lmax[h], l);
    }
  }
#pragma unroll
  for (int h = 0; h < H; h++) {
    red[tid] = lmax[h];
    __syncthreads();
    for (int off = 32; off > 0; off >>= 1) {
      if (tid < off) red[tid] = fmaxf(red[tid], red[tid + off]);
      __syncthreads();
    }
    if (tid == 0) mh[h] = red[0];
    __syncthreads();
  }
  float lsum[H];
#pragma unroll
  for (int h = 0; h < H; h++) lsum[h] = 0.f;
  for (int pos = start + tid; pos < end; pos += T) {
#pragma unroll
    for (int h = 0; h < H; h++) {
      float e = expf(wbuf[pos * H + h] - mh[h]);
      wbuf[pos * H + h] = e;
      lsum[h] += e;
    }
  }
#pragma unroll
  for (int h = 0; h < H; h++) {
    red[tid] = lsum[h];
    __syncthreads();
    for (int off = 32; off > 0; off >>= 1) {
      if (tid < off) red[tid] += red[tid + off];
      __syncthreads();
    }
    if (tid == 0) sh[h] = red[0];
    __syncthreads();
  }
  for (int pos = start + tid; pos < end; pos += T) {
#pragma unroll
    for (int h = 0; h < H; h++) wbuf[pos * H + h] /= (sh[h] + 1e-16f);
  }
}

// weighted gather-aggregate: out[n,:] = elu( sum_e w[e,h] * hlin[src_e,:] + b )
template <int H>
__global__ void aggr_kernel(const int* __restrict__ rowptr, const int* __restrict__ csr_src,
                            const float* __restrict__ wbuf, const float* __restrict__ hlin,
                            const float* __restrict__ bias, float* __restrict__ out) {
  const int CO = H * 64;
  int n = blockIdx.x;
  int tid = threadIdx.x;  // blockDim == CO
  int h = tid >> 6;
  int start = rowptr[n], end = rowptr[n + 1];
  float acc = 0.f;
  for (int pos = start; pos < end; ++pos) {
    int s = csr_src[pos];
    if (pos + 1 < end) __builtin_prefetch(&hlin[(long)csr_src[pos + 1] * CO + tid], 0, 1);
    float w = wbuf[pos * H + h];
    acc += hlin[(long)s * CO + tid] * w;
  }
  float v = acc + bias[tid];
  v = (v > 0.f) ? v : (expf(v) - 1.f);  // elu
  out[(long)n * CO + tid] = v;
}

// ---------------------------------------------------------------- readout + MLP
__global__ void batchsum_kernel(const float* __restrict__ h3, const int* __restrict__ batch,
                                float* gsum, float* gcnt) {
  int n = blockIdx.x;
  int tid = threadIdx.x;  // 64
  int g = batch[n];
  atomicAdd(&gsum[g * 64 + tid], h3[(long)n * 64 + tid]);
  if (tid == 0) atomicAdd(&gcnt[g], 1.0f);
}

__global__ void mlp_kernel(const float* gsum, const float* gcnt, const float* p1w,
                           const float* p1b, const float* p2w, const float* p2b,
                           const float* p3w, const float* p3b, float* dout, int G) {
  int g = blockIdx.x;
  int tid = threadIdx.x;  // 64
  __shared__ float r[64], z1[64], z2[32];
  float inv = 1.0f / fmaxf(gcnt[g], 1.0f);
  float rv = gsum[g * 64 + tid] * inv;
  r[tid] = rv;
  dout[G + g * 64 + tid] = rv;  // readout output (second tuple element)
  __syncthreads();
  float s = p1b[tid];
  for (int k = 0; k < 64; k++) s += r[k] * p1w[k * 64 + tid];
  z1[tid] = fmaxf(s, 0.f);
  __syncthreads();
  if (tid < 32) {
    float s2 = p2b[tid];
    for (int k = 0; k < 64; k++) s2 += z1[k] * p2w[k * 32 + tid];
    z2[tid] = fmaxf(s2, 0.f);
  }
  __syncthreads();
  if (tid == 0) {
    float o = p3b[0];
    for (int k = 0; k < 32; k++) o += z2[k] * p3w[k];
    dout[g] = o;
  }
}

// ---------------------------------------------------------------- driver
extern "C" void kernel_launch(void* const* d_in, const int* in_sizes, int n_in,
                              void* d_out, int out_size, void* d_ws, size_t ws_size,
                              hipStream_t stream) {
  const float* x    = (const float*)d_in[0];
  const int* ei     = (const int*)d_in[1];
  const float* ea   = (const float*)d_in[2];
  const int* batch  = (const int*)d_in[3];
  const float* W1   = (const float*)d_in[4];
  const float* We1  = (const float*)d_in[5];
  const float* as1  = (const float*)d_in[6];
  const float* ad1  = (const float*)d_in[7];
  const float* ae1  = (const float*)d_in[8];
  const float* b1   = (const float*)d_in[9];
  const float* W2   = (const float*)d_in[10];
  const float* We2  = (const float*)d_in[11];
  const float* as2  = (const float*)d_in[12];
  const float* ad2  = (const float*)d_in[13];
  const float* ae2  = (const float*)d_in[14];
  const float* b2   = (const float*)d_in[15];
  const float* W3   = (const float*)d_in[16];
  const float* We3  = (const float*)d_in[17];
  const float* as3  = (const float*)d_in[18];
  const float* ad3  = (const float*)d_in[19];
  const float* ae3  = (const float*)d_in[20];
  const float* b3   = (const float*)d_in[21];
  const float* p1w  = (const float*)d_in[22];
  const float* p1b  = (const float*)d_in[23];
  const float* p2w  = (const float*)d_in[24];
  const float* p2b  = (const float*)d_in[25];
  const float* p3w  = (const float*)d_in[26];
  const float* p3b  = (const float*)d_in[27];

  const int N = in_sizes[3];
  const int E = in_sizes[1] / 2;
  const int Etot = E + N;
  const int G = out_size / 65;  // out_size = G + G*64
  const int* srcp = ei;
  const int* dstp = ei + E;

  // ---- workspace layout
  char* ws = (char*)d_ws;
  size_t off = 0;
  auto take = [&](size_t bytes) -> char* {
    char* p = ws + off;
    off = (off + bytes + 255) & ~(size_t)255;
    return p;
  };
  // zero-initialized region first
  int*   cnt       = (int*)take((size_t)N * 4);
  int*   fill      = (int*)take((size_t)N * 4);
  float* loop_attr = (float*)take((size_t)N * 6 * 4);
  float* gsum      = (float*)take((size_t)G * 64 * 4);
  float* gcnt      = (float*)take((size_t)G * 4);
  size_t zero_bytes = off;
  // rest
  int*   rowptr  = (int*)take((size_t)(N + 1) * 4);
  int*   csr_src = (int*)take((size_t)Etot * 4);
  float* ea_csr  = (float*)take((size_t)Etot * 6 * 4);
  float* wbuf    = (float*)take((size_t)Etot * 4 * 4);
  float* sbuf    = (float*)take((size_t)N * 4 * 4);
  float* dbuf    = (float*)take((size_t)N * 4 * 4);
  float* we_ae   = (float*)take(64 * 4);
  float* actA    = (float*)take((size_t)N * 256 * 4);
  float* actB    = (float*)take((size_t)N * 256 * 4);
  (void)ws_size; (void)n_in;

  // ---- build CSR (+ self loops, loop_attr mean)
  long zero_words = (long)(zero_bytes / 4);
  zero_ws_kernel<<<2048, 256, 0, stream>>>((unsigned int*)ws, zero_words);
  count_kernel<<<(E + 255) / 256, 256, 0, stream>>>(dstp, ea, cnt, loop_attr, E);
  finalize_kernel<<<(N + 255) / 256, 256, 0, stream>>>(cnt, loop_attr, N);
  scan_kernel<<<1, 256, 0, stream>>>(cnt, rowptr, N);
  fill_edges_kernel<<<(E + 255) / 256, 256, 0, stream>>>(srcp, dstp, ea, rowptr, fill, csr_src,
                                                         ea_csr, E);
  fill_loops_kernel<<<(N + 255) / 256, 256, 0, stream>>>(loop_attr, rowptr, fill, csr_src,
                                                         ea_csr, N);
  weae_kernel<<<1, 64, 0, stream>>>(We1, ae1, We2, ae2, We3, ae3, we_ae);

  // ---- layer 1 (K=7 GEMM, H=4, CO=256)
  gemm7_kernel<<<N, 256, 0, stream>>>(x, W1, actA, N);
  score_kernel<4><<<(N * 4 + 255) / 256, 256, 0, stream>>>(actA, as1, ad1, sbuf, dbuf, N);
  attn_kernel<4><<<N, 64, 0, stream>>>(rowptr, csr_src, ea_csr, sbuf, dbuf, we_ae + 0, wbuf);
  aggr_kernel<4><<<N, 256, 0, stream>>>(rowptr, csr_src, wbuf, actA, b1, actB);

  // ---- layer 2 (WMMA fp32 GEMM 256x256, H=4)
  gemm_wmma_kernel<256><<<(N + 31) / 32, 256, 0, stream>>>(actB, W2, actA, N);
  score_kernel<4><<<(N * 4 + 255) / 256, 256, 0, stream>>>(actA, as2, ad2, sbuf, dbuf, N);
  attn_kernel<4><<<N, 64, 0, stream>>>(rowptr, csr_src, ea_csr, sbuf, dbuf, we_ae + 24, wbuf);
  aggr_kernel<4><<<N, 256, 0, stream>>>(rowptr, csr_src, wbuf, actA, b2, actB);

  // ---- layer 3 (WMMA fp32 GEMM 256x64, H=1)
  gemm_wmma_kernel<64><<<(N + 127) / 128, 256, 0, stream>>>(actB, W3, actA, N);
  score_kernel<1><<<(N + 255) / 256, 256, 0, stream>>>(actA, as3, ad3, sbuf, dbuf, N);
  attn_kernel<1><<<N, 64, 0, stream>>>(rowptr, csr_src, ea_csr, sbuf, dbuf, we_ae + 48, wbuf);
  aggr_kernel<1><<<N, 64, 0, stream>>>(rowptr, csr_src, wbuf, actA, b3, actB);

  // ---- readout + MLP
  batchsum_kernel<<<N, 64, 0, stream>>>(actB, batch, gsum, gcnt);
  mlp_kernel<<<G, 64, 0, stream>>>(gsum, gcnt, p1w, p1b, p2w, p2b, p3w, p3b, (float*)d_out, G);
}